// MinibatchDiscrimination_74646531604915
// MI455X (gfx1250) — compile-verified
//
#include <hip/hip_runtime.h>
#include <hip/hip_bf16.h>
#include <math.h>

// Problem constants (from reference)
#define BATCH    1024
#define FEAT     512
#define NKERN    50
#define KDIM     5
#define NCOL     (NKERN * KDIM)   // 250
#define NCOL_PAD 256              // padded to 16 tiles of 16
#define OUTF     (FEAT + NKERN)   // 562

typedef __attribute__((ext_vector_type(2))) float        v2f;
typedef __attribute__((ext_vector_type(8))) float        v8f;
typedef __attribute__((ext_vector_type(4))) unsigned int v4u;
typedef __attribute__((ext_vector_type(4))) int          v4i;
typedef __attribute__((ext_vector_type(8))) int          v8i;

// Tensor Data Mover availability (ROCm 7.2 clang-22: 5-arg; clang-23: 6-arg)
#if defined(__has_builtin)
#  if __has_builtin(__builtin_amdgcn_tensor_load_to_lds) && \
      __has_builtin(__builtin_amdgcn_s_wait_tensorcnt)
#    define HAVE_TDM 1
#  endif
#endif
#ifndef HAVE_TDM
#  define HAVE_TDM 0
#endif

// ---------------------------------------------------------------------------
// Kernel 0: transpose + zero-pad T into Tt[256][512] (column-major of T).
// Removes all bounds guards from the GEMM hot loop and makes the B fragment
// a single contiguous b64 load. 128K elements -> negligible.
// ---------------------------------------------------------------------------
__global__ void __launch_bounds__(256)
transpose_pad_T(const float* __restrict__ T,   // [FEAT][NCOL]
                float* __restrict__ Tt)        // [NCOL_PAD][FEAT]
{
    const int c = blockIdx.x;                  // 0..255 (padded column)
    const bool ok = (c < NCOL);
    for (int k = threadIdx.x; k < FEAT; k += 256)
        Tt[c * FEAT + k] = ok ? T[k * NCOL + c] : 0.0f;
}

// ---------------------------------------------------------------------------
// Kernel 1: M = x @ T via V_WMMA_F32_16X16X4_F32 (full fp32 precision).
// One wave per 16x16 output tile. Grid: (1024/16, 256/16) = (64, 16).
//
// ISA 32-bit A-matrix 16x4 layout: lane L holds row M = L&15;
//   VGPR0 = K = (L>>4)*2, VGPR1 = K+1.
// B 4x16 mirrors: lane L holds col N = L&15; VGPR0 = row K=(L>>4)*2, VGPR1=K+1.
// C/D 16x16 f32: VGPR v, lanes 0-15 -> row v, lanes 16-31 -> row v+8.
// Branch-free loop: EXEC stays all-ones as WMMA requires.
// ---------------------------------------------------------------------------
__global__ void __launch_bounds__(32)
gemm_wmma_f32(const float* __restrict__ A,    // [BATCH][FEAT]
              const float* __restrict__ Tt,   // [NCOL_PAD][FEAT] (transposed)
              float* __restrict__ M)          // [BATCH][NCOL_PAD] (workspace)
{
    const int lane  = threadIdx.x & 31;
    const int i0    = blockIdx.x * 16;          // row tile base
    const int n0    = blockIdx.y * 16;          // col tile base
    const int row   = i0 + (lane & 15);         // A row this lane serves
    const int col   = n0 + (lane & 15);         // B col this lane serves
    const int khalf = (lane >> 4) * 2;          // 0 or 2

    const float* ap = A  + row * FEAT + khalf;  // A[row][khalf + k]
    const float* bp = Tt + col * FEAT + khalf;  // T[khalf + k][col] transposed

    v8f acc = {};
#pragma unroll 4
    for (int k = 0; k < FEAT; k += 4) {
        v2f a;                                   // one global_load_b64
        a.x = ap[k];
        a.y = ap[k + 1];
        v2f b;                                   // one global_load_b64
        b.x = bp[k];
        b.y = bp[k + 1];
        // 8 args: (neg_a, A, neg_b, B, c_mod, C, reuse_a, reuse_b)
        acc = __builtin_amdgcn_wmma_f32_16x16x4_f32(
            false, a, false, b, (short)0, acc, false, false);
    }

    // Store D: lane L writes col n0+(L&15); VGPR v -> row i0 + v + 8*(L>=16)
    const int rbase = i0 + ((lane >> 4) << 3);
    const int c     = n0 + (lane & 15);
#pragma unroll
    for (int v = 0; v < 8; ++v) {
        M[(rbase + v) * NCOL_PAD + c] = acc[v];
    }
}

// ---------------------------------------------------------------------------
// Kernel 2: copy x into out[:, 0:512]. One block per row, float2 moves
// (562 is even -> 8-byte aligned row starts on both sides).
// ---------------------------------------------------------------------------
__global__ void __launch_bounds__(256)
copy_x(const float* __restrict__ x, float* __restrict__ out)
{
    const int row = blockIdx.x;
    const float2* src = reinterpret_cast<const float2*>(x + row * FEAT);
    float2*       dst = reinterpret_cast<float2*>(out + row * OUTF);
    dst[threadIdx.x] = src[threadIdx.x];   // 256 float2 = 512 floats
}

// ---------------------------------------------------------------------------
// Kernel 3: minibatch features. Grid (BATCH/256, NKERN), block 256 (8 waves).
// Stage M[:, k, :] (1024 x 5 floats = 20 KB) into LDS via the Tensor Data
// Mover (one 2D D#: tile 5 x 1024, row stride 256 elems, 4B elements,
// compacted contiguously into LDS == sM's row-major layout), then every
// thread owns one i and loops all j. LDS reads are wave-broadcast.
// exp(-s) lowers to a single v_exp_f32 via __expf.
// ---------------------------------------------------------------------------
__global__ void __launch_bounds__(256)
pairwise(const float* __restrict__ M,   // [BATCH][NCOL_PAD]
         float* __restrict__ out)       // [BATCH][OUTF]
{
    const int k = blockIdx.y;
    const int i = blockIdx.x * 256 + threadIdx.x;

    __shared__ float sM[BATCH * KDIM];  // 20 KB of the WGP's 320 KB

#if HAVE_TDM
    if (threadIdx.x < 32) {             // wave 0 issues the DMA (EXEC ignored)
        const unsigned long long ga =
            (unsigned long long)(uintptr_t)(M + k * KDIM);   // tile start VA
        const unsigned lds_off = (unsigned)(uintptr_t)(&sM[0]); // low 32 = LDS byte addr

        // D# group 0: count=1 | lds_addr | global_addr[56:0] | type=2
        v4u g0;
        g0.x = 1u;
        g0.y = lds_off;
        g0.z = (unsigned)(ga & 0xFFFFFFFFu);
        g0.w = (unsigned)((ga >> 32) & 0x1FFFFFFu) | (2u << 30);

        // D# group 1 (bit layout per CDNA5 ISA):
        v8i g1;
        g1[0] = (int)(2u << 16);        // workgroup_mask=0, data_size=2 (4B)
        g1[1] = (int)(256u << 16);      // tensor_dim0 = 256 (elems per line)
        g1[2] = (int)(1024u << 16);     // tensor_dim0 hi=0 | tensor_dim1 = 1024
        g1[3] = (int)(5u << 16);        // tensor_dim1 hi=0 | tile_dim0 = 5
        g1[4] = 1024;                   // tile_dim1 = 1024, tile_dim2 = 0
        g1[5] = 256;                    // tensor_dim0_stride = 256 elems
        g1[6] = 0;                      // stride hi | tensor_dim1_stride lo
        g1[7] = 0;                      // tensor_dim1_stride hi

        v4i gz = {};                    // groups 2/3 unused (2D tensor)
#if __clang_major__ >= 23
        v8i gz8 = {};
        __builtin_amdgcn_tensor_load_to_lds(g0, g1, gz, gz, gz8, 0);
#else
        __builtin_amdgcn_tensor_load_to_lds(g0, g1, gz, gz, 0);
#endif
        __builtin_amdgcn_s_wait_tensorcnt(0);   // TENSORcnt -> 0
    }
    __syncthreads();
#else
    // Fallback: cooperative fill (thread t loads rows t, t+256, t+512, t+768)
    for (int j = threadIdx.x; j < BATCH; j += 256) {
        const float* mp = M + j * NCOL_PAD + k * KDIM;
#pragma unroll
        for (int d = 0; d < KDIM; ++d)
            sM[j * KDIM + d] = mp[d];
    }
    __syncthreads();
#endif

    const float m0 = sM[i * KDIM + 0];
    const float m1 = sM[i * KDIM + 1];
    const float m2 = sM[i * KDIM + 2];
    const float m3 = sM[i * KDIM + 3];
    const float m4 = sM[i * KDIM + 4];

    float acc = 0.0f;
#pragma unroll 4
    for (int j = 0; j < BATCH; ++j) {
        const float* mj = &sM[j * KDIM];
        float s = fabsf(m0 - mj[0]);
        s += fabsf(m1 - mj[1]);
        s += fabsf(m2 - mj[2]);
        s += fabsf(m3 - mj[3]);
        s += fabsf(m4 - mj[4]);
        acc += __expf(-s);              // v_mul + v_exp_f32
    }

    out[i * OUTF + FEAT + k] = acc;
}

// ---------------------------------------------------------------------------
extern "C" void kernel_launch(void* const* d_in, const int* in_sizes, int n_in,
                              void* d_out, int out_size, void* d_ws, size_t ws_size,
                              hipStream_t stream)
{
    const float* x = (const float*)d_in[0];   // [1024, 512] fp32
    const float* T = (const float*)d_in[1];   // [512, 250] fp32
    float* out = (float*)d_out;               // [1024, 562] fp32

    // Workspace layout: M [1024][256] (1 MB) then Tt [256][512] (512 KB)
    float* M  = (float*)d_ws;
    float* Tt = M + BATCH * NCOL_PAD;

    // 0) Tt = pad(T)^T  (branch-free GEMM loads)
    transpose_pad_T<<<NCOL_PAD, 256, 0, stream>>>(T, Tt);

    // 1) M = x @ T on the WMMA path
    gemm_wmma_f32<<<dim3(BATCH / 16, NCOL_PAD / 16), 32, 0, stream>>>(x, Tt, M);

    // 2) out[:, :512] = x
    copy_x<<<BATCH, 256, 0, stream>>>(x, out);

    // 3) out[:, 512:562] = minibatch features (reads M, ordered on stream)
    pairwise<<<dim3(BATCH / 256, NKERN), 256, 0, stream>>>(M, out);
}